// CompressedModel_88888643158215
// MI455X (gfx1250) — compile-verified
//
#include <hip/hip_runtime.h>
#include <math.h>

#ifndef M_PI
#define M_PI 3.14159265358979323846
#endif

typedef __attribute__((ext_vector_type(2))) float v2f;
typedef __attribute__((ext_vector_type(8))) float v8f;

#define BATCH 64
#define TDIM  1024   // token length (K of GEMM1)
#define DDIM  768    // hidden dim (N of both GEMMs)
#define KKEEP 922    // ceil(0.9 * 1024)
#define MPAD  960    // 15 * 64, covers 922 rows with 64-wide wave tiles
#define K2PAD 928    // 922 rounded up to multiple of 4 (and 16)

// ---------------------------------------------------------------------------
// Build orthonormal DCT-II matrix C1[kr, t] (922x1024 logical, zero-padded to
// 960x1024).  C1[kr,t] = s(kr) * cos(pi*(2t+1)*kr / (2*1024))
// ---------------------------------------------------------------------------
__global__ __launch_bounds__(256) void build_dct1(float* __restrict__ C1) {
    int idx = blockIdx.x * 256 + threadIdx.x;
    if (idx >= MPAD * TDIM) return;
    int t  = idx & (TDIM - 1);
    int kr = idx >> 10;
    float v = 0.0f;
    if (kr < KKEEP) {
        double s   = (kr == 0) ? sqrt(1.0 / (double)TDIM) : sqrt(2.0 / (double)TDIM);
        double ang = M_PI * ((2.0 * (double)t + 1.0) * (double)kr) / (2.0 * (double)TDIM);
        v = (float)(s * cos(ang));
    }
    C1[idx] = v;
}

// ---------------------------------------------------------------------------
// Build C2[n, m] = C_k[m, n]  (transpose of orthonormal DCT-II of size 922),
// zero-padded to 960 x 928.  C_k[m,n] = s(m) * cos(pi*(2n+1)*m / (2*922))
// ---------------------------------------------------------------------------
__global__ __launch_bounds__(256) void build_dct2(float* __restrict__ C2) {
    int idx = blockIdx.x * 256 + threadIdx.x;
    if (idx >= MPAD * K2PAD) return;
    int m = idx % K2PAD;
    int n = idx / K2PAD;
    float v = 0.0f;
    if (n < KKEEP && m < KKEEP) {
        double s   = (m == 0) ? sqrt(1.0 / (double)KKEEP) : sqrt(2.0 / (double)KKEEP);
        double ang = M_PI * ((2.0 * (double)n + 1.0) * (double)m) / (2.0 * (double)KKEEP);
        v = (float)(s * cos(ang));
    }
    C2[idx] = v;
}

// ---------------------------------------------------------------------------
// Batched fp32 WMMA GEMM:  D[b] = A @ B[b]
//   A:  [>=Mtiles*64 x lda] row-major, zero-padded (shared across batch)
//   B:  [Kvalid x DDIM] row-major per batch (stride strideB)
//   D:  [Mvalid x DDIM] row-major per batch (stride strideD)
// One wave (32 threads) computes a 64x64 tile via 4x4 V_WMMA_F32_16X16X4_F32
// accumulators.  Kmain is a multiple of 4; if Kvalid > Kmain, one guarded
// tail step handles the ragged K edge with zero-fill (no branchy loads: the
// address is clamped and the value masked, so EXEC stays all-ones for WMMA).
// ---------------------------------------------------------------------------
__global__ __launch_bounds__(32) void wmma_gemm_f32(
    const float* __restrict__ A, int lda,
    const float* __restrict__ Bg, unsigned long long strideB,
    float* __restrict__ Dg, unsigned long long strideD,
    int Mvalid, int Kmain, int Kvalid)
{
    const int N = DDIM;
    int lane  = threadIdx.x;
    int lrow  = lane & 15;   // M (for A), N (for B/D) position within 16-tile
    int khalf = lane >> 4;   // 0: K rows 0,1 ; 1: K rows 2,3 (per ISA layout)

    int m0 = blockIdx.x * 64;
    int n0 = blockIdx.y * 64;
    const float* __restrict__ Bb = Bg + strideB * (unsigned long long)blockIdx.z;
    float* __restrict__ Db       = Dg + strideD * (unsigned long long)blockIdx.z;

    v8f acc[4][4];
#pragma unroll
    for (int mt = 0; mt < 4; ++mt)
#pragma unroll
        for (int nt = 0; nt < 4; ++nt)
            acc[mt][nt] = (v8f){0.f,0.f,0.f,0.f,0.f,0.f,0.f,0.f};

    const float* Arow[4];
#pragma unroll
    for (int mt = 0; mt < 4; ++mt)
        Arow[mt] = A + (size_t)(m0 + mt * 16 + lrow) * (size_t)lda + khalf * 2;

    for (int k0 = 0; k0 < Kmain; k0 += 4) {
        v2f af[4];
#pragma unroll
        for (int mt = 0; mt < 4; ++mt)
            af[mt] = *(const v2f*)(Arow[mt] + k0);          // K=khalf*2, khalf*2+1

        int kr = k0 + khalf * 2;
        const float* b0p = Bb + (size_t)kr * N + n0 + lrow;
        v2f bf[4];
#pragma unroll
        for (int nt = 0; nt < 4; ++nt) {
            bf[nt].x = b0p[nt * 16];                         // row kr
            bf[nt].y = b0p[nt * 16 + N];                     // row kr+1
        }

#pragma unroll
        for (int mt = 0; mt < 4; ++mt)
#pragma unroll
            for (int nt = 0; nt < 4; ++nt)
                acc[mt][nt] = __builtin_amdgcn_wmma_f32_16x16x4_f32(
                    false, af[mt], false, bf[nt],
                    (short)0, acc[mt][nt], false, false);
    }

    if (Kvalid > Kmain) {   // ragged K tail (GEMM2: K rows 920..921 valid)
        int kr  = Kmain + khalf * 2;
        bool ok0 = kr < Kvalid;
        bool ok1 = (kr + 1) < Kvalid;
        int  r0  = ok0 ? kr : 0;
        int  r1  = ok1 ? (kr + 1) : 0;

        v2f af[4];
#pragma unroll
        for (int mt = 0; mt < 4; ++mt)
            af[mt] = *(const v2f*)(Arow[mt] + Kmain);        // A zero-padded in K

        v2f bf[4];
#pragma unroll
        for (int nt = 0; nt < 4; ++nt) {
            int col  = n0 + nt * 16 + lrow;
            float x0 = Bb[(size_t)r0 * N + col];
            float x1 = Bb[(size_t)r1 * N + col];
            bf[nt].x = ok0 ? x0 : 0.0f;
            bf[nt].y = ok1 ? x1 : 0.0f;
        }

#pragma unroll
        for (int mt = 0; mt < 4; ++mt)
#pragma unroll
            for (int nt = 0; nt < 4; ++nt)
                acc[mt][nt] = __builtin_amdgcn_wmma_f32_16x16x4_f32(
                    false, af[mt], false, bf[nt],
                    (short)0, acc[mt][nt], false, false);
    }

    // Store: VGPR v of a 16x16 C/D tile holds row (v + khalf*8), col = lrow.
#pragma unroll
    for (int mt = 0; mt < 4; ++mt) {
        int rbase = m0 + mt * 16 + khalf * 8;
#pragma unroll
        for (int nt = 0; nt < 4; ++nt) {
            int col = n0 + nt * 16 + lrow;
#pragma unroll
            for (int v = 0; v < 8; ++v) {
                int r = rbase + v;
                if (r < Mvalid)
                    Db[(size_t)r * N + col] = acc[mt][nt][v];
            }
        }
    }
}

// ---------------------------------------------------------------------------
extern "C" void kernel_launch(void* const* d_in, const int* in_sizes, int n_in,
                              void* d_out, int out_size, void* d_ws, size_t ws_size,
                              hipStream_t stream) {
    (void)in_sizes; (void)n_in; (void)out_size; (void)ws_size;

    const float* x = (const float*)d_in[0];          // [64,1024,768] fp32
    float* out = (float*)d_out;

    // Workspace layout: C1 [960x1024], then C2 [960x928]
    float* C1 = (float*)d_ws;
    float* C2 = C1 + (size_t)MPAD * TDIM;

    const size_t rec_elems = (size_t)BATCH * KKEEP * DDIM;
    float* x_rec = out;                  // output 0
    float* x_dct = out + rec_elems;      // output 1

    build_dct1<<<(MPAD * TDIM  + 255) / 256, 256, 0, stream>>>(C1);
    build_dct2<<<(MPAD * K2PAD + 255) / 256, 256, 0, stream>>>(C2);

    dim3 grid((MPAD + 63) / 64, DDIM / 64, BATCH);   // (15, 12, 64)

    // GEMM1: x_dct[b] = C1(922x1024) @ x[b](1024x768)
    wmma_gemm_f32<<<grid, 32, 0, stream>>>(
        C1, TDIM,
        x, (unsigned long long)TDIM * DDIM,
        x_dct, (unsigned long long)KKEEP * DDIM,
        KKEEP, TDIM, TDIM);

    // GEMM2: x_rec[b] = C2(922x922, Kpad 928) @ x_dct[b](922x768)
    wmma_gemm_f32<<<grid, 32, 0, stream>>>(
        C2, K2PAD,
        x_dct, (unsigned long long)KKEEP * DDIM,
        x_rec, (unsigned long long)KKEEP * DDIM,
        KKEEP, KKEEP & ~3, KKEEP);
}